// IrisSpecializedLoss_12695923327132
// MI455X (gfx1250) — compile-verified
//
#include <hip/hip_runtime.h>
#include <math.h>

// Problem constants (fixed by reference).
#define NUM_C 10
#define H_    30
#define W_    30
#define HW    (H_ * W_)

typedef __attribute__((ext_vector_type(2))) float v2f;
typedef __attribute__((ext_vector_type(8))) float v8f;

// Wave32 cross-lane reduction of TWO scalars using V_WMMA_F32_16X16X4_F32
// (ones/selector-matrix trick).
//
// A (16x4 f32) layout: lane l, vgpr v -> A[l%16][v + 2*(l/16)].
//   a = {p0, p1}  =>  A[m][0]=p0_m, A[m][1]=p1_m (lanes 0..15)
//                     A[m][2]=p0_{m+16}, A[m][3]=p1_{m+16} (lanes 16..31)
// B (4x16 f32) layout: lane l, vgpr v -> B[v + 2*(l/16)][l%16].
//   Choose B[0][n]=B[2][n]=(n<8), B[1][n]=B[3][n]=(n>=8):
//     D[m][n<8]  = p0_m + p0_{m+16}
//     D[m][n>=8] = p1_m + p1_{m+16}
// Each lane sums its 8 D registers (rows 0-7 or 8-15 of its column), then one
// shfl_xor(16) completes the 32-lane total: columns 0-7 carry sum(p0),
// columns 8-15 carry sum(p1). EXEC must be all-ones (uniform call sites only).
__device__ __forceinline__ void waveReducePair(float p0, float p1,
                                               float& s0, float& s1) {
  const int lane = threadIdx.x & 31;
  v2f a; a.x = p0; a.y = p1;
  const float selLo = ((lane & 8) == 0) ? 1.0f : 0.0f;
  v2f b; b.x = selLo; b.y = 1.0f - selLo;
  v8f c = {};
  c = __builtin_amdgcn_wmma_f32_16x16x4_f32(
      /*neg_a=*/false, a, /*neg_b=*/false, b,
      /*c_mod=*/(short)0, c, /*reuse_a=*/false, /*reuse_b=*/false);
  float s = c[0] + c[1] + c[2] + c[3] + c[4] + c[5] + c[6] + c[7];
  s += __shfl_xor(s, 16, 32);
  s0 = __shfl(s, 0, 32);   // any lane with (lane&8)==0 holds sum(p0)
  s1 = __shfl(s, 8, 32);   // any lane with (lane&8)!=0 holds sum(p1)
}

// One block per batch element. 256 threads stream 900 pixels.
// Writes 5 partials per block to part[b*8 + {0..4}]:
//   0: sum (logsumexp - x[t])        (cross-entropy numerator)
//   1: sum (att^2) - 2*sum att[t]    (color-mapping numerator, minus N)
//   2: sum eq * windowWeight(h,w)    (consistency numerator)
//   3: sum eq                        (exact count)
//   4: per-batch rule-loss sum of squares
__global__ void iris_block_loss(const float* __restrict__ logits,
                                const int*   __restrict__ targets,
                                const float* __restrict__ att,
                                const float* __restrict__ lstm,
                                float* __restrict__ part) {
  const int b   = blockIdx.x;
  const int tid = threadIdx.x;

  __shared__ int   s_pres;
  __shared__ float s_wave[8][4];
  if (tid == 0) s_pres = 0;
  __syncthreads();

  const float* lg = logits + (size_t)b * NUM_C * HW;
  const int*   tg = targets + (size_t)b * HW;
  const float* at = att + (size_t)b * HW * NUM_C;

  float ce = 0.f, cm = 0.f, eqw = 0.f, cnt = 0.f;
  int mask = 0;

  for (int p = tid; p < HW; p += blockDim.x) {
    const int t = tg[p];
    mask |= (1 << t);

    // ---- logits: coalesced across threads for each channel ----
    float x[NUM_C];
#pragma unroll
    for (int c = 0; c < NUM_C; ++c) x[c] = lg[c * HW + p];

    float m = x[0]; int am = 0;
#pragma unroll
    for (int c = 1; c < NUM_C; ++c) { if (x[c] > m) { m = x[c]; am = c; } }
    float se = 0.f;
#pragma unroll
    for (int c = 0; c < NUM_C; ++c) se += __expf(x[c] - m);
    ce += m + __logf(se) - x[t];

    const float e = (am == t) ? 1.0f : 0.0f;
    cnt += e;
    const int h = p / W_, w = p - h * W_;
    const float wgt = ((h == 0 || h == H_ - 1) ? 1.f : 2.f) *
                      ((w == 0 || w == W_ - 1) ? 1.f : 2.f);
    eqw += e * wgt;

    // ---- attention: 10 contiguous floats, 8B aligned -> 5x b64 loads ----
    float av[NUM_C];
    const float2* ap = reinterpret_cast<const float2*>(at + (size_t)p * NUM_C);
#pragma unroll
    for (int c2 = 0; c2 < NUM_C / 2; ++c2) {
      float2 v = ap[c2];
      av[2 * c2] = v.x; av[2 * c2 + 1] = v.y;
    }
    float s2 = 0.f, avt = 0.f;
#pragma unroll
    for (int c = 0; c < NUM_C; ++c) {
      s2 += av[c] * av[c];
      if (c == t) avt = av[c];
    }
    cm += s2 - 2.f * avt;
  }

  atomicOr(&s_pres, mask);

  // WMMA-based wave reductions (uniform control flow, EXEC all ones).
  float t0, t1, t2, t3;
  waveReducePair(ce,  cm,  t0, t1);
  waveReducePair(eqw, cnt, t2, t3);
  const int wave = tid >> 5;
  if ((tid & 31) == 0) {
    s_wave[wave][0] = t0; s_wave[wave][1] = t1;
    s_wave[wave][2] = t2; s_wave[wave][3] = t3;
  }
  __syncthreads();

  if (tid == 0) {
    float a0 = 0.f, a1 = 0.f, a2 = 0.f, a3 = 0.f;
    const int nw = blockDim.x >> 5;
    for (int i = 0; i < nw; ++i) {
      a0 += s_wave[i][0]; a1 += s_wave[i][1];
      a2 += s_wave[i][2]; a3 += s_wave[i][3];
    }
    // Per-batch rules: sorted unique present colors into leading slots.
    float rules[NUM_C];
    for (int c = 0; c < NUM_C; ++c) rules[c] = 0.f;
    int slot = 0;
    const int pm = s_pres;
    for (int c = 0; c < NUM_C; ++c)
      if (pm & (1 << c)) rules[slot++] = (float)c;
    float rl = 0.f;
    const float* lr = lstm + (size_t)b * NUM_C;
    for (int j = 0; j < NUM_C; ++j) { float d = lr[j] - rules[j]; rl += d * d; }

    float* o = part + (size_t)b * 8;
    o[0] = a0; o[1] = a1; o[2] = a2; o[3] = a3; o[4] = rl;
  }
}

// Single-block finalize: sum 4096 block records, compute the 6 outputs.
__global__ void iris_finalize(const float* __restrict__ part,
                              float* __restrict__ out, int numBlocks) {
  const int tid = threadIdx.x;
  float s0 = 0.f, s1 = 0.f, s2 = 0.f, s3 = 0.f, s4 = 0.f;
  for (int i = tid; i < numBlocks; i += blockDim.x) {
    const float* p = part + (size_t)i * 8;
    s0 += p[0]; s1 += p[1]; s2 += p[2]; s3 += p[3]; s4 += p[4];
  }
  float t0, t1, t2, t3, t4, tz;
  waveReducePair(s0, s1, t0, t1);
  waveReducePair(s2, s3, t2, t3);
  waveReducePair(s4, 0.f, t4, tz);

  __shared__ float sw[8][5];
  const int wave = tid >> 5;
  if ((tid & 31) == 0) {
    sw[wave][0] = t0; sw[wave][1] = t1; sw[wave][2] = t2;
    sw[wave][3] = t3; sw[wave][4] = t4;
  }
  __syncthreads();

  if (tid == 0) {
    float a0 = 0.f, a1 = 0.f, a2 = 0.f, a3 = 0.f, a4 = 0.f;
    const int nw = blockDim.x >> 5;
    for (int i = 0; i < nw; ++i) {
      a0 += sw[i][0]; a1 += sw[i][1]; a2 += sw[i][2];
      a3 += sw[i][3]; a4 += sw[i][4];
    }
    const float B = (float)numBlocks;
    const float N = B * (float)HW;
    const float ce   = a0 / N;
    const float cmap = (a1 + N) / (N * (float)NUM_C);
    const float cons = 1.f - a2 / (B * (float)((H_ - 1) * (W_ - 1)) * 4.f);
    const float rl   = a4 / (B * (float)NUM_C);
    const float cnt  = a3;
    float total = ce + 0.4f * cmap + 0.3f * cons + 0.2f * rl
                  - (cnt / B) * 5.0f;
    total = fmaxf(total, 0.001f);
    out[0] = total; out[1] = ce; out[2] = cmap;
    out[3] = cons;  out[4] = rl; out[5] = cnt;
  }
}

extern "C" void kernel_launch(void* const* d_in, const int* in_sizes, int n_in,
                              void* d_out, int out_size, void* d_ws, size_t ws_size,
                              hipStream_t stream) {
  const float* logits  = (const float*)d_in[0];  // [B,10,30,30] f32
  const int*   targets = (const int*)  d_in[1];  // [B,30,30]    i32
  const float* att     = (const float*)d_in[2];  // [B,30,30,10] f32
  const float* lstm    = (const float*)d_in[3];  // [B,10]       f32
  const int B = in_sizes[1] / HW;                // 4096

  float* part = (float*)d_ws;                    // B * 8 floats (128 KB)

  iris_block_loss<<<B, 256, 0, stream>>>(logits, targets, att, lstm, part);
  iris_finalize<<<1, 256, 0, stream>>>(part, (float*)d_out, B);
}